// ZoeDepthAttractorLayerUnnormed_55052890800771
// MI455X (gfx1250) — compile-verified
//
#include <hip/hip_runtime.h>
#include <math.h>

typedef __attribute__((ext_vector_type(2))) float v2f;
typedef __attribute__((ext_vector_type(8))) float v8f;

#define IN_F   256
#define MLP    128
#define NATT   16
#define NBINS  64
#define PTILE  64      // pixels per block
#define H      128
#define W      128
#define SRC    64      // source resolution of emb / b_prev
#define NB     4       // batch

__device__ __forceinline__ float softplus_stable(float v) {
    // logaddexp(v, 0) = max(v,0) + log1p(exp(-|v|))
    return fmaxf(v, 0.0f) + log1pf(__expf(-fabsf(v)));
}

__global__ __launch_bounds__(256)
void attractor_fused(const float* __restrict__ x,
                     const float* __restrict__ b_prev,
                     const float* __restrict__ emb,
                     const float* __restrict__ w1,
                     const float* __restrict__ b1,
                     const float* __restrict__ w2,
                     const float* __restrict__ b2,
                     float* __restrict__ out)
{
    __shared__ float s_xin [PTILE * IN_F];   // [pixel][channel]  64 KB
    __shared__ float s_hidd[PTILE * MLP];    // [pixel][hidden]   32 KB
    __shared__ float s_att [PTILE * NATT];   // [pixel][attractor] 4 KB

    const int tid = threadIdx.x;
    const int blk = blockIdx.x;
    const int ox_base = (blk & 1) * PTILE;       // 2 segments per row
    const int oy      = (blk >> 1) & (H - 1);
    const int n       = blk >> 8;

    // align_corners=True scale for both axes (128 -> 64 source)
    const float sc = (float)(SRC - 1) / (float)(H - 1);
    const float fy = (float)oy * sc;
    const int   iy0 = (int)fy;
    const int   iy1 = min(iy0 + 1, SRC - 1);
    const float wy  = fy - (float)iy0;

    // ---- Stage 1: xin[p][c] = x + bilinear(prev_b_embedding) ----
    for (int idx = tid; idx < IN_F * PTILE; idx += 256) {
        const int c = idx >> 6;             // idx / PTILE
        const int p = idx & (PTILE - 1);
        const int ox = ox_base + p;
        const float fx = (float)ox * sc;
        const int   ix0 = (int)fx;
        const int   ix1 = min(ix0 + 1, SRC - 1);
        const float wx  = fx - (float)ix0;

        const float* eb = emb + (size_t)(n * IN_F + c) * SRC * SRC;
        const float e00 = eb[iy0 * SRC + ix0];
        const float e01 = eb[iy0 * SRC + ix1];
        const float e10 = eb[iy1 * SRC + ix0];
        const float e11 = eb[iy1 * SRC + ix1];
        const float ev  = (1.f - wy) * ((1.f - wx) * e00 + wx * e01)
                        +        wy  * ((1.f - wx) * e10 + wx * e11);
        const float xv = x[((size_t)(n * IN_F + c) * H + oy) * W + ox];
        s_xin[p * IN_F + c] = xv + ev;
    }
    __syncthreads();

    const int lane = tid & 31;
    const int wave = tid >> 5;      // 0..7
    const int ln16 = lane & 15;
    const int hi   = lane >> 4;     // 0: K0/K1 half, 1: K2/K3 half

    // ---- Stage 2: GEMM1  hidd = relu(W1 * xin + b1), f32 WMMA 16x16x4 ----
    {
        const int m_base = wave * 16;                       // 8 waves cover MLP=128
        v8f acc0 = {}, acc1 = {}, acc2 = {}, acc3 = {};
        const float* w1row = w1 + (size_t)(m_base + ln16) * IN_F + hi * 2;
        const float* bp0 = &s_xin[( 0 + ln16) * IN_F + hi * 2];
        const float* bp1 = &s_xin[(16 + ln16) * IN_F + hi * 2];
        const float* bp2 = &s_xin[(32 + ln16) * IN_F + hi * 2];
        const float* bp3 = &s_xin[(48 + ln16) * IN_F + hi * 2];
        #pragma unroll 4
        for (int k = 0; k < IN_F; k += 4) {
            const v2f a  = *(const v2f*)(w1row + k);
            const v2f bb0 = *(const v2f*)(bp0 + k);
            const v2f bb1 = *(const v2f*)(bp1 + k);
            const v2f bb2 = *(const v2f*)(bp2 + k);
            const v2f bb3 = *(const v2f*)(bp3 + k);
            acc0 = __builtin_amdgcn_wmma_f32_16x16x4_f32(false, a, false, bb0, (short)0, acc0, false, false);
            acc1 = __builtin_amdgcn_wmma_f32_16x16x4_f32(false, a, false, bb1, (short)0, acc1, false, false);
            acc2 = __builtin_amdgcn_wmma_f32_16x16x4_f32(false, a, false, bb2, (short)0, acc2, false, false);
            acc3 = __builtin_amdgcn_wmma_f32_16x16x4_f32(false, a, false, bb3, (short)0, acc3, false, false);
        }
        #pragma unroll
        for (int i = 0; i < 8; ++i) {
            const int m = m_base + hi * 8 + i;          // D row = vgpr + 8*(lane>=16)
            const float bias = b1[m];
            s_hidd[( 0 + ln16) * MLP + m] = fmaxf(acc0[i] + bias, 0.f);
            s_hidd[(16 + ln16) * MLP + m] = fmaxf(acc1[i] + bias, 0.f);
            s_hidd[(32 + ln16) * MLP + m] = fmaxf(acc2[i] + bias, 0.f);
            s_hidd[(48 + ln16) * MLP + m] = fmaxf(acc3[i] + bias, 0.f);
        }
    }
    __syncthreads();

    // ---- Stage 3: GEMM2  A = softplus(W2 * hidd + b2) ----
    if (wave < 4) {
        const int pix = wave * 16;
        v8f acc = {};
        const float* w2row = w2 + (size_t)(ln16) * MLP + hi * 2;
        const float* bp = &s_hidd[(pix + ln16) * MLP + hi * 2];
        #pragma unroll 4
        for (int k = 0; k < MLP; k += 4) {
            const v2f a = *(const v2f*)(w2row + k);
            const v2f b = *(const v2f*)(bp + k);
            acc = __builtin_amdgcn_wmma_f32_16x16x4_f32(false, a, false, b, (short)0, acc, false, false);
        }
        #pragma unroll
        for (int i = 0; i < 8; ++i) {
            const int aix = hi * 8 + i;
            s_att[(pix + ln16) * NATT + aix] = softplus_stable(acc[i] + b2[aix]);
        }
    }
    __syncthreads();

    // ---- Stage 4: b_centers bilinear + attractor pull + output (x2 tuple) ----
    const size_t out_half = (size_t)NB * NBINS * H * W;
    for (int idx = tid; idx < NBINS * PTILE; idx += 256) {
        const int p = idx & (PTILE - 1);
        const int k = idx >> 6;
        const int ox = ox_base + p;
        const float fx = (float)ox * sc;
        const int   ix0 = (int)fx;
        const int   ix1 = min(ix0 + 1, SRC - 1);
        const float wx  = fx - (float)ix0;

        const float* bp = b_prev + (size_t)(n * NBINS + k) * SRC * SRC;
        const float c00 = bp[iy0 * SRC + ix0];
        const float c01 = bp[iy0 * SRC + ix1];
        const float c10 = bp[iy1 * SRC + ix0];
        const float c11 = bp[iy1 * SRC + ix1];
        const float bc  = (1.f - wy) * ((1.f - wx) * c00 + wx * c01)
                        +        wy  * ((1.f - wx) * c10 + wx * c11);

        float delta = 0.f;
        #pragma unroll
        for (int a = 0; a < NATT; ++a) {
            const float dx = s_att[p * NATT + a] - bc;
            delta += dx / (1.0f + 300.0f * dx * dx);    // inv_attractor, gamma=2
        }
        const float r = bc + delta;
        const size_t o = ((size_t)(n * NBINS + k) * H + oy) * W + ox;
        out[o] = r;
        out[o + out_half] = r;
    }
}

extern "C" void kernel_launch(void* const* d_in, const int* in_sizes, int n_in,
                              void* d_out, int out_size, void* d_ws, size_t ws_size,
                              hipStream_t stream) {
    const float* x      = (const float*)d_in[0];
    const float* b_prev = (const float*)d_in[1];
    const float* emb    = (const float*)d_in[2];
    const float* w1     = (const float*)d_in[3];
    const float* b1     = (const float*)d_in[4];
    const float* w2     = (const float*)d_in[5];
    const float* b2     = (const float*)d_in[6];
    float* out = (float*)d_out;

    // one block per (n, row, 64-pixel segment): 4 * 128 * 2 = 1024 blocks, 8 waves each
    dim3 grid(NB * H * (W / PTILE));
    attractor_fused<<<grid, 256, 0, stream>>>(x, b_prev, emb, w1, b1, w2, b2, out);
}